// InVol_54898271977622
// MI455X (gfx1250) — compile-verified
//
#include <hip/hip_runtime.h>

// f32 WMMA fragment types: A = 16x4 f32 (2 VGPRs), C/D = 16x16 f32 (8 VGPRs)
typedef __attribute__((ext_vector_type(2))) float v2f;
typedef __attribute__((ext_vector_type(8))) float v8f;

#define TS       16            // spatial tile (16x16 interior pixels)
#define HCOLS    18            // halo tile extent
#define XSTRIDE  20            // padded x-dim stride within a channel row
#define XROW     1296          // floats per halo y-row = 64*20 + 16 (bank-stagger pad)
#define NPIX     256           // pixels per tile
#define PSTRIDE  266           // padded pixel stride for ys/ks (bank-conflict free)
#define WRS      68            // padded w_reduce row stride (64 -> 68)
#define WSPS     36            // padded w_span row stride (32 -> 36)

#define SMEM_FLOATS (18 * XROW + 32 * PSTRIDE + 80 * PSTRIDE + 32 * WRS + 80 * WSPS + 32 + 80)

__device__ __forceinline__ v8f wmma4(v2f a, v2f b, v8f c) {
  // D = A(16x4,f32) * B(4x16,f32) + C(16x16,f32)
  return __builtin_amdgcn_wmma_f32_16x16x4_f32(
      /*neg_a=*/false, a, /*neg_b=*/false, b,
      /*c_mod=*/(short)0, c, /*reuse_a=*/false, /*reuse_b=*/false);
}

__global__ __launch_bounds__(256) void InVol_54898271977622_kernel(
    const float* __restrict__ x, const float* __restrict__ w_reduce,
    const float* __restrict__ b_reduce, const float* __restrict__ w_span,
    const float* __restrict__ b_span, float* __restrict__ out) {
  extern __shared__ float smem[];
  float* xs  = smem;                    // halo tile of x: [y][c][x], 18*1296
  float* ys  = xs + 18 * XROW;          // Y  : [32][PSTRIDE]
  float* ks  = ys + 32 * PSTRIDE;       // kern: [80][PSTRIDE] (rows 72..79 = 0)
  float* wr  = ks + 80 * PSTRIDE;       // w_reduce: [32][WRS]
  float* wsp = wr + 32 * WRS;           // w_span (padded to 80 rows): [80][WSPS]
  float* br  = wsp + 80 * WSPS;         // b_reduce[32]
  float* bsp = br + 32;                 // b_span padded[80]

  const int t  = threadIdx.x;
  const int bi = blockIdx.z;
  const int h0 = blockIdx.y * TS;
  const int w0 = blockIdx.x * TS;

  // ---------------- cooperative tile + weight load ----------------
  for (int idx = t; idx < 64 * HCOLS * HCOLS; idx += 256) {
    int c   = idx / (HCOLS * HCOLS);
    int rem = idx - c * (HCOLS * HCOLS);
    int yy  = rem / HCOLS;
    int xx  = rem - yy * HCOLS;
    int gy = h0 + yy - 1, gx = w0 + xx - 1;
    float v = 0.0f;
    if (gy >= 0 && gy < 128 && gx >= 0 && gx < 128)
      v = x[((bi * 64 + c) * 128 + gy) * 128 + gx];
    xs[yy * XROW + c * XSTRIDE + xx] = v;
  }
  for (int idx = t; idx < 2048; idx += 256)
    wr[(idx >> 6) * WRS + (idx & 63)] = w_reduce[idx];
  for (int idx = t; idx < 2560; idx += 256) {
    int m = idx >> 5, k = idx & 31;
    wsp[m * WSPS + k] = (m < 72) ? w_span[m * 32 + k] : 0.0f;
  }
  if (t < 32) br[t] = b_reduce[t];
  if (t < 80) bsp[t] = (t < 72) ? b_span[t] : 0.0f;
  __syncthreads();

  const int wv = t >> 5;       // wave id 0..7 -> owns pixels [32*wv, 32*wv+32)
  const int lane = t & 31;
  const int lh = lane >> 4;    // lane half (ISA fragment layout)
  const int ln = lane & 15;

  // ------- stage 1: Y[32 x Npix] = Wr[32x64] * X[64 x Npix], bias in C-init -------
  v8f accY[2][2];
#pragma unroll
  for (int mt = 0; mt < 2; ++mt)
#pragma unroll
    for (int nt = 0; nt < 2; ++nt)
#pragma unroll
      for (int i = 0; i < 8; ++i)
        accY[mt][nt][i] = br[mt * 16 + i + 8 * lh];

#pragma unroll
  for (int kb = 0; kb < 64; kb += 4) {
    v2f a[2], b[2];
#pragma unroll
    for (int mt = 0; mt < 2; ++mt) {
      const float* p = &wr[(mt * 16 + ln) * WRS + kb + 2 * lh];
      v2f av = {p[0], p[1]};             // A vgpr0:K=0|2, vgpr1:K=1|3 per lane half
      a[mt] = av;
    }
#pragma unroll
    for (int nt = 0; nt < 2; ++nt) {
      int ty = 2 * wv + nt;              // pixel row of this N-tile
      const float* p = &xs[(ty + 1) * XROW + (kb + 2 * lh) * XSTRIDE + ln + 1];
      v2f bv = {p[0], p[XSTRIDE]};       // B: K rows striped across lane halves
      b[nt] = bv;
    }
#pragma unroll
    for (int mt = 0; mt < 2; ++mt)
#pragma unroll
      for (int nt = 0; nt < 2; ++nt)
        accY[mt][nt] = wmma4(a[mt], b[nt], accY[mt][nt]);
  }
  // scatter C-layout accumulators into ys[row][pixel] (same-wave columns only)
#pragma unroll
  for (int mt = 0; mt < 2; ++mt)
#pragma unroll
    for (int nt = 0; nt < 2; ++nt) {
      int p = (2 * wv + nt) * 16 + ln;
#pragma unroll
      for (int i = 0; i < 8; ++i)
        ys[(mt * 16 + i + 8 * lh) * PSTRIDE + p] = accY[mt][nt][i];
    }
  // No __syncthreads(): ys/ks columns for this wave's pixels are produced and
  // consumed within the same wave; DScnt ordering suffices.

  // ------- stage 2: kern[80 x Npix] = Wsp[80x32] * Y, bias in C-init -------
  for (int nt = 0; nt < 2; ++nt) {
    int p = (2 * wv + nt) * 16 + ln;
    v8f accK[5];
#pragma unroll
    for (int mt = 0; mt < 5; ++mt)
#pragma unroll
      for (int i = 0; i < 8; ++i)
        accK[mt][i] = bsp[mt * 16 + i + 8 * lh];
#pragma unroll
    for (int kb = 0; kb < 32; kb += 4) {
      v2f bv = {ys[(kb + 2 * lh) * PSTRIDE + p],
                ys[(kb + 2 * lh + 1) * PSTRIDE + p]};
#pragma unroll
      for (int mt = 0; mt < 5; ++mt) {
        const float* q = &wsp[(mt * 16 + ln) * WSPS + kb + 2 * lh];
        v2f av = {q[0], q[1]};
        accK[mt] = wmma4(av, bv, accK[mt]);
      }
    }
#pragma unroll
    for (int mt = 0; mt < 5; ++mt)
#pragma unroll
      for (int i = 0; i < 8; ++i)
        ks[(mt * 16 + i + 8 * lh) * PSTRIDE + p] = accK[mt][i];
  }

  // ------- stage 3: out[g*8+c'] = sum_k kern[g,k] * x[c', 3x3 nbhd] -------
  {
    int p = t;                      // one pixel per thread (same wave as producer)
    int ty = p >> 4, tx = p & 15;
#pragma unroll 1
    for (int g = 0; g < 8; ++g) {
      float kv[9];
#pragma unroll
      for (int k = 0; k < 9; ++k)
        kv[k] = ks[(g * 9 + k) * PSTRIDE + p];
#pragma unroll
      for (int cc = 0; cc < 8; ++cc) {
        int c = g * 8 + cc;
        const float* xb = &xs[ty * XROW + c * XSTRIDE + tx];
        float s = 0.0f;
#pragma unroll
        for (int k = 0; k < 9; ++k)
          s = fmaf(kv[k], xb[(k / 3) * XROW + (k % 3)], s);
        out[((bi * 64 + c) * 128 + (h0 + ty)) * 128 + (w0 + tx)] = s;
      }
    }
  }
}

extern "C" void kernel_launch(void* const* d_in, const int* in_sizes, int n_in,
                              void* d_out, int out_size, void* d_ws, size_t ws_size,
                              hipStream_t stream) {
  const float* x        = (const float*)d_in[0];
  const float* w_reduce = (const float*)d_in[1];
  const float* b_reduce = (const float*)d_in[2];
  const float* w_span   = (const float*)d_in[3];
  const float* b_span   = (const float*)d_in[4];
  float* out = (float*)d_out;

  const size_t smem = (size_t)SMEM_FLOATS * sizeof(float);   // ~228 KB < 320 KB/WGP
  (void)hipFuncSetAttribute((const void*)InVol_54898271977622_kernel,
                            hipFuncAttributeMaxDynamicSharedMemorySize, (int)smem);

  dim3 grid(128 / TS, 128 / TS, 8);   // (w tiles, h tiles, batch) = 8x8x8
  InVol_54898271977622_kernel<<<grid, 256, smem, stream>>>(
      x, w_reduce, b_reduce, w_span, b_span, out);
}